// my_network_70136815944258
// MI455X (gfx1250) — compile-verified
//
#include <hip/hip_runtime.h>
#include <math.h>

// ---------------------------------------------------------------------------
// JPEG-sim pipeline for MI455X (gfx1250), fp32 WMMA path, v3.
//  - 16x16 pixel tile per wave32; block-diagonal DCT diag(A,A) -> one 16x16
//    matmul does four independent 8x8 DCTs. V_WMMA_F32_16X16X4_F32 chained x4.
//  - LDS staging, row stride 20 floats: bank-conflict-free strided A reads
//    AND 16B-aligned rows -> b128 global/LDS traffic in pixel phases.
//  - Quant {q,1/q} pairs hoisted to registers once per wave (no per-channel
//    table reloads, no runtime division).
//  - Persistent waves (4 tiles each) + global_prefetch of next tile.
//  - Per-wave s_wait_dscnt fences (LDS is in-order per wave) -> no s_barrier.
// ---------------------------------------------------------------------------

#define IMG_H 2048
#define IMG_W 2048
#define HW (IMG_H * IMG_W)
#define TILES_X (IMG_W / 16)    // 128
#define TILES_Y (IMG_H / 16)    // 128
#define NTILES (TILES_X * TILES_Y)
#define WAVES 8
#define NBLOCKS 512
#define GRID_WAVES (NBLOCKS * WAVES)  // 4096 waves -> 4 tiles per wave
#define TS 20                   // LDS row stride: conflict-free + 16B aligned

typedef float v2f __attribute__((ext_vector_type(2)));
typedef float v4f __attribute__((ext_vector_type(4)));
typedef float v8f __attribute__((ext_vector_type(8)));

// {q, 1/q} pairs, compile-time exact reciprocals (Q_QUALITY=50 => scale 1.0)
#define P(x) {x##.0f, 1.0f / x##.0f}
__constant__ v2f c_QP[2][64] = {
  { // luma
    P(16),P(11),P(10),P(16),P(24),P(40),P(51),P(61),
    P(12),P(12),P(14),P(19),P(26),P(58),P(60),P(55),
    P(14),P(13),P(16),P(24),P(40),P(57),P(69),P(56),
    P(14),P(17),P(22),P(29),P(51),P(87),P(80),P(62),
    P(18),P(22),P(37),P(56),P(68),P(109),P(103),P(77),
    P(24),P(35),P(55),P(64),P(81),P(104),P(113),P(92),
    P(49),P(64),P(78),P(87),P(103),P(121),P(120),P(101),
    P(72),P(92),P(95),P(98),P(112),P(100),P(103),P(99)
  },
  { // chroma
    P(17),P(18),P(24),P(47),P(99),P(99),P(99),P(99),
    P(18),P(21),P(26),P(66),P(99),P(99),P(99),P(99),
    P(24),P(26),P(56),P(99),P(99),P(99),P(99),P(99),
    P(47),P(66),P(99),P(99),P(99),P(99),P(99),P(99),
    P(99),P(99),P(99),P(99),P(99),P(99),P(99),P(99),
    P(99),P(99),P(99),P(99),P(99),P(99),P(99),P(99),
    P(99),P(99),P(99),P(99),P(99),P(99),P(99),P(99),
    P(99),P(99),P(99),P(99),P(99),P(99),P(99),P(99)
  }
};
#undef P

__device__ __forceinline__ float dct8(int i, int j) {
  float a = (i == 0) ? 0.35355339059327373f : 0.5f;
  return a * cosf(3.14159265358979323846f * ((float)j + 0.5f) * (float)i * 0.125f);
}
// 16x16 block-diagonal DCT matrix diag(A, A)
__device__ __forceinline__ float ablk16(int i, int j) {
  return ((i >> 3) == (j >> 3)) ? dct8(i & 7, j & 7) : 0.0f;
}
// LDS ops are in-order per wave; DS drain + compiler memory fence.
__device__ __forceinline__ void lds_fence() {
  asm volatile("s_wait_dscnt 0" ::: "memory");
}
__device__ __forceinline__ v8f wmma4(v2f a, v2f b, v8f c) {
  // (neg_a, A, neg_b, B, c_mod, C, reuse_a, reuse_b)
  return __builtin_amdgcn_wmma_f32_16x16x4_f32(false, a, false, b, (short)0, c,
                                               false, false);
}

// One channel: X -> Ablk*X*Ablk^T -> quant-sim -> Ablk^T*(.)*Ablk, in place.
__device__ __forceinline__ void dct_pipeline(float* tile, float* scr,
                                             const v2f* F1, const v2f* F2,
                                             const float* q, const float* rq,
                                             int lo, int hi) {
  // mm1: M1 = Ablk * X          (A = const F1, B from tile)
  v8f m1 = {};
#pragma unroll
  for (int kc = 0; kc < 4; ++kc) {
    int kb = 4 * kc + 2 * hi;
    v2f b; b.x = tile[kb * TS + lo]; b.y = tile[(kb + 1) * TS + lo];
    m1 = wmma4(F1[kc], b, m1);
  }
#pragma unroll
  for (int j = 0; j < 8; ++j) scr[(j + 8 * hi) * TS + lo] = m1[j];
  lds_fence();

  // mm2: D = M1 * Ablk^T        (A from scr, B = const F1)
  v8f d = {};
#pragma unroll
  for (int kc = 0; kc < 4; ++kc) {
    int kb = 4 * kc + 2 * hi;
    v2f a; a.x = scr[lo * TS + kb]; a.y = scr[lo * TS + kb + 1];
    d = wmma4(a, F1[kc], d);
  }

  // quantization simulation: t=D/Q; r=rne(t); u=(r+(t-r)^3)*Q
  v8f u;
#pragma unroll
  for (int j = 0; j < 8; ++j) {
    float tt = d[j] * rq[j];
    float r  = rintf(tt);          // round-to-nearest-even == jnp.round
    float e  = tt - r;
    u[j] = (r + e * e * e) * q[j];
  }
#pragma unroll
  for (int j = 0; j < 8; ++j) scr[(j + 8 * hi) * TS + lo] = u[j];
  lds_fence();

  // mm3: R1 = Ablk^T * U        (A = const F2, B from scr)
  v8f r1 = {};
#pragma unroll
  for (int kc = 0; kc < 4; ++kc) {
    int kb = 4 * kc + 2 * hi;
    v2f b; b.x = scr[kb * TS + lo]; b.y = scr[(kb + 1) * TS + lo];
    r1 = wmma4(F2[kc], b, r1);
  }
#pragma unroll
  for (int j = 0; j < 8; ++j) scr[(j + 8 * hi) * TS + lo] = r1[j];
  lds_fence();

  // mm4: R = R1 * Ablk          (A from scr, B = const F2)
  v8f rec = {};
#pragma unroll
  for (int kc = 0; kc < 4; ++kc) {
    int kb = 4 * kc + 2 * hi;
    v2f a; a.x = scr[lo * TS + kb]; a.y = scr[lo * TS + kb + 1];
    rec = wmma4(a, F2[kc], rec);
  }
#pragma unroll
  for (int j = 0; j < 8; ++j) tile[(j + 8 * hi) * TS + lo] = rec[j];
  lds_fence();
}

__global__ __launch_bounds__(WAVES * 32)
void jpeg_sim_kernel(const float* __restrict__ in, float* __restrict__ out) {
  __shared__ float lds[WAVES][4][16 * TS];  // [wave][Y,Cb,Cr,scratch]

  const int lane = threadIdx.x & 31;
  const int wv   = threadIdx.x >> 5;
  const int lo   = lane & 15;
  const int hi   = (lane >> 4) & 1;

  // --- constant DCT operand fragments (one-time) ---
  // F1: Ablk as A-frag (mm1)  == Ablk^T as B-frag (mm2)
  // F2: Ablk^T as A-frag (mm3) == Ablk as B-frag (mm4)
  v2f F1[4], F2[4];
#pragma unroll
  for (int kc = 0; kc < 4; ++kc) {
    int kb = 4 * kc + 2 * hi;
    v2f f1; f1.x = ablk16(lo, kb); f1.y = ablk16(lo, kb + 1); F1[kc] = f1;
    v2f f2; f2.x = ablk16(kb, lo); f2.y = ablk16(kb + 1, lo); F2[kc] = f2;
  }

  // --- quant tables to registers, once per wave ---
  float qY[8], rqY[8], qC[8], rqC[8];
#pragma unroll
  for (int j = 0; j < 8; ++j) {
    v2f p0 = c_QP[0][j * 8 + (lane & 7)];
    v2f p1 = c_QP[1][j * 8 + (lane & 7)];
    qY[j] = p0.x; rqY[j] = p0.y;
    qC[j] = p1.x; rqC[j] = p1.y;
  }

  float* tY  = lds[wv][0];
  float* tU  = lds[wv][1];
  float* tV  = lds[wv][2];
  float* scr = lds[wv][3];

  const int gw = blockIdx.x * WAVES + wv;  // global wave id, 0..4095

#pragma unroll 1
  for (int t = gw; t < NTILES; t += GRID_WAVES) {
    const int th = t >> 7;                  // /TILES_X
    const int tw = t & (TILES_X - 1);
    const int h0 = th * 16, w0 = tw * 16;

    // --- prefetch next tile (48 cache lines: 16 rows x 3 channels) ---
    {
      int tn = t + GRID_WAVES; if (tn >= NTILES) tn = t;
      const int hn = (tn >> 7) * 16, wn = (tn & (TILES_X - 1)) * 16;
      const float* pf = in + (size_t)hi * HW + (size_t)(hn + lo) * IMG_W + wn;
      __builtin_prefetch(pf, 0, 1);          // ch0 (lanes 0-15) / ch1 (16-31)
      if (lane < 16) __builtin_prefetch(pf + 2 * (size_t)HW, 0, 1);  // ch2
    }

    // --- Phase A: b128 loads, forward color transform, stage YCbCr ---
#pragma unroll
    for (int i = 0; i < 2; ++i) {
      int qi = i * 32 + lane;               // quad index, 64 quads per tile
      int r = qi >> 2, c4 = (qi & 3) * 4;
      size_t g = (size_t)(h0 + r) * IMG_W + (w0 + c4);
      v4f R = *(const v4f*)(in + g);
      v4f G = *(const v4f*)(in + HW + g);
      v4f B = *(const v4f*)(in + 2 * (size_t)HW + g);
      v4f Y =  0.299f * R + 0.587f * G + 0.114f * B;
      v4f U = -0.168736f * R - 0.331264f * G + 0.5f * B + 128.0f;
      v4f V =  0.5f * R - 0.418688f * G - 0.081312f * B + 128.0f;
      *(v4f*)(tY + r * TS + c4) = Y;
      *(v4f*)(tU + r * TS + c4) = U;
      *(v4f*)(tV + r * TS + c4) = V;
    }
    lds_fence();

    // --- Phase B: luma then 2x chroma (register-resident quant tables) ---
    dct_pipeline(tY, scr, F1, F2, qY, rqY, lo, hi);
#pragma unroll 1
    for (int ch = 1; ch < 3; ++ch)
      dct_pipeline(lds[wv][ch], scr, F1, F2, qC, rqC, lo, hi);

    // --- Phase C: inverse color (JFIF), normalize, b128 stores ---
#pragma unroll
    for (int i = 0; i < 2; ++i) {
      int qi = i * 32 + lane;
      int r = qi >> 2, c4 = (qi & 3) * 4;
      v4f Y  = *(const v4f*)(tY + r * TS + c4);
      v4f Cb = *(const v4f*)(tU + r * TS + c4) - 128.0f;
      v4f Cr = *(const v4f*)(tV + r * TS + c4) - 128.0f;
      v4f R = Y + 1.402f * Cr;
      v4f G = Y - 0.344136f * Cb - 0.714136f * Cr;
      v4f B = Y + 1.772f * Cb;
      size_t g = (size_t)(h0 + r) * IMG_W + (w0 + c4);
      // (x/255 - mean)/std  ==  (x - 255*mean) / (255*std)
      *(v4f*)(out + g)                  = (R - 125.3f) * (1.0f / 63.0f);
      *(v4f*)(out + HW + g)             = (G - 123.0f) * (1.0f / 62.1f);
      *(v4f*)(out + 2 * (size_t)HW + g) = (B - 113.9f) * (1.0f / 66.7f);
    }
    lds_fence();  // tile buffers reused next iteration
  }
}

extern "C" void kernel_launch(void* const* d_in, const int* in_sizes, int n_in,
                              void* d_out, int out_size, void* d_ws, size_t ws_size,
                              hipStream_t stream) {
  const float* in = (const float*)d_in[0];
  float* out = (float*)d_out;
  dim3 grid(NBLOCKS);              // 512 workgroups, persistent
  dim3 block(WAVES * 32);          // 8 wave32 per workgroup
  hipLaunchKernelGGL(jpeg_sim_kernel, grid, block, 0, stream, in, out);
}